// GCN_70093866270993
// MI455X (gfx1250) — compile-verified
//
#include <hip/hip_runtime.h>
#include <cmath>

// ---------------------------------------------------------------------------
// GCN forward on gfx1250 (MI455X). Dense x@W1 uses V_WMMA_F32_16X16X4_F32 with
// the B operand (W1, zero-padded to 16 columns) staged in LDS once per block.
// ---------------------------------------------------------------------------

typedef __attribute__((ext_vector_type(2))) float v2f;
typedef __attribute__((ext_vector_type(8))) float v8f;

static __host__ int cdiv(int a, int b) { return (a + b - 1) / b; }

// -------------------- degree / normalization --------------------
__global__ void k_deg_init(float* __restrict__ deg, int n) {
  int i = blockIdx.x * blockDim.x + threadIdx.x;
  if (i < n) deg[i] = 1.0f;  // self-loop contributes 1 to every node's degree
}

__global__ void k_deg_accum(const int* __restrict__ dst, float* __restrict__ deg, int e) {
  int i = blockIdx.x * blockDim.x + threadIdx.x;
  if (i < e) unsafeAtomicAdd(&deg[dst[i]], 1.0f);
}

__global__ void k_deg_to_dis(float* __restrict__ deg, int n) {
  int i = blockIdx.x * blockDim.x + threadIdx.x;
  if (i < n) deg[i] = rsqrtf(deg[i]);  // deg >= 1 always (self loops)
}

// -------------------- layer-1 GEMM: [N,128] @ [128,4] via WMMA --------------------
// One wave computes a 16-row tile. A = 16x4 fp32 slice of x, B = 4x16 (W1 columns
// 0..3, zero-padded to 16 columns, staged transposed in LDS as Bs[n][k]).
// K loop: 128/4 = 32 WMMA f32_16x16x4 accumulations.
// Lane layouts per CDNA5 ISA 7.12.2:
//   A (16x4, 2 VGPR): lanes 0-15 -> M=lane, K={0,1}; lanes 16-31 -> M=lane-16, K={2,3}
//   B (4x16, 2 VGPR): lanes 0-15 -> N=lane, K={0,1}; lanes 16-31 -> N=lane-16, K={2,3}
//   C/D (16x16, 8 VGPR): VGPR r: lanes 0-15 -> M=r, N=lane; lanes 16-31 -> M=r+8, N=lane-16
__global__ void k_gemm1_wmma(const float* __restrict__ x, const float* __restrict__ W1,
                             float* __restrict__ out, int n) {
  __shared__ float Bs[16 * 128];  // Bs[n*128 + k] = (n < 4) ? W1[k*4 + n] : 0

  // Cooperative, branch-light LDS fill (blockDim.x == 128 -> 16 entries/thread).
  for (int idx = threadIdx.x; idx < 16 * 128; idx += blockDim.x) {
    int col = idx >> 7;     // n (output column), 0..15
    int k   = idx & 127;    // K index
    Bs[idx] = (col < 4) ? W1[k * 4 + col] : 0.0f;
  }
  __syncthreads();

  const int lane = threadIdx.x & 31;
  const int waveInBlock = threadIdx.x >> 5;
  const int waveId = blockIdx.x * (blockDim.x >> 5) + waveInBlock;
  const int rowBase = waveId * 16;
  if (rowBase >= n) return;  // wave-uniform: EXEC stays all-ones for WMMA

  const int m  = lane & 15;        // row-in-tile (A) / column (B, C/D)
  const int kh = (lane >> 4) * 2;  // K sub-offset: 0 for lanes 0-15, 2 for lanes 16-31

  const float* __restrict__ xrow = x + (size_t)(rowBase + m) * 128;
  const float* __restrict__ blane = Bs + m * 128 + kh;  // contiguous K pairs, 8B aligned

  v8f acc = {};
#pragma unroll
  for (int k0 = 0; k0 < 128; k0 += 4) {
    v2f a = *reinterpret_cast<const v2f*>(xrow + k0 + kh);   // global_load_b64
    v2f b = *reinterpret_cast<const v2f*>(blane + k0);       // ds_load_b64
    acc = __builtin_amdgcn_wmma_f32_16x16x4_f32(
        /*neg_a=*/false, a, /*neg_b=*/false, b,
        /*c_mod=*/(short)0, acc, /*reuse_a=*/false, /*reuse_b=*/false);
  }

  // Write back columns 0..3 of the 16x16 tile.
  if (m < 4) {
    const int rshift = (lane >> 4) * 8;  // lanes 16-31 hold rows 8..15
#pragma unroll
    for (int r = 0; r < 8; ++r) {
      out[(size_t)(rowBase + rshift + r) * 4 + m] = acc[r];
    }
  }
}

// -------------------- aggregation (self-loop init + edge scatter) --------------------
__global__ void k_self4(const float* __restrict__ dis, const float* __restrict__ in,
                        float* __restrict__ out, int n) {
  int i = blockIdx.x * blockDim.x + threadIdx.x;
  if (i >= n) return;
  float w = dis[i] * dis[i];
  float4 v = reinterpret_cast<const float4*>(in)[i];
  reinterpret_cast<float4*>(out)[i] = make_float4(v.x * w, v.y * w, v.z * w, v.w * w);
}

__global__ void k_self2(const float* __restrict__ dis, const float* __restrict__ in,
                        float* __restrict__ out, int n) {
  int i = blockIdx.x * blockDim.x + threadIdx.x;
  if (i >= n) return;
  float w = dis[i] * dis[i];
  float2 v = reinterpret_cast<const float2*>(in)[i];
  reinterpret_cast<float2*>(out)[i] = make_float2(v.x * w, v.y * w);
}

__global__ void k_edge4(const int* __restrict__ src, const int* __restrict__ dst,
                        const float* __restrict__ dis, const float* __restrict__ in,
                        float* __restrict__ out, int e) {
  int i = blockIdx.x * blockDim.x + threadIdx.x;
  if (i >= e) return;
  int s = src[i], d = dst[i];
  float w = dis[s] * dis[d];
  float4 v = reinterpret_cast<const float4*>(in)[s];
  float* o = out + (size_t)d * 4;
  unsafeAtomicAdd(o + 0, v.x * w);
  unsafeAtomicAdd(o + 1, v.y * w);
  unsafeAtomicAdd(o + 2, v.z * w);
  unsafeAtomicAdd(o + 3, v.w * w);
}

__global__ void k_edge2(const int* __restrict__ src, const int* __restrict__ dst,
                        const float* __restrict__ dis, const float* __restrict__ in,
                        float* __restrict__ out, int e) {
  int i = blockIdx.x * blockDim.x + threadIdx.x;
  if (i >= e) return;
  int s = src[i], d = dst[i];
  float w = dis[s] * dis[d];
  float2 v = reinterpret_cast<const float2*>(in)[s];
  float* o = out + (size_t)d * 2;
  unsafeAtomicAdd(o + 0, v.x * w);
  unsafeAtomicAdd(o + 1, v.y * w);
}

// -------------------- fused activation + tiny GEMM --------------------
// h = tanh(agg + b) [N,4]; out = h @ W (4x4 row-major)
__global__ void k_act_gemm4x4(const float* __restrict__ agg, const float* __restrict__ b,
                              const float* __restrict__ W, float* __restrict__ out, int n) {
  int i = blockIdx.x * blockDim.x + threadIdx.x;
  if (i >= n) return;
  float4 v = reinterpret_cast<const float4*>(agg)[i];
  float h0 = tanhf(v.x + b[0]);
  float h1 = tanhf(v.y + b[1]);
  float h2 = tanhf(v.z + b[2]);
  float h3 = tanhf(v.w + b[3]);
  float4 o;
  o.x = h0 * W[0] + h1 * W[4] + h2 * W[8]  + h3 * W[12];
  o.y = h0 * W[1] + h1 * W[5] + h2 * W[9]  + h3 * W[13];
  o.z = h0 * W[2] + h1 * W[6] + h2 * W[10] + h3 * W[14];
  o.w = h0 * W[3] + h1 * W[7] + h2 * W[11] + h3 * W[15];
  reinterpret_cast<float4*>(out)[i] = o;
}

// h = tanh(agg + b) [N,4]; out = h @ W (4x2 row-major) -> [N,2]
__global__ void k_act_gemm4x2(const float* __restrict__ agg, const float* __restrict__ b,
                              const float* __restrict__ W, float* __restrict__ out, int n) {
  int i = blockIdx.x * blockDim.x + threadIdx.x;
  if (i >= n) return;
  float4 v = reinterpret_cast<const float4*>(agg)[i];
  float h0 = tanhf(v.x + b[0]);
  float h1 = tanhf(v.y + b[1]);
  float h2 = tanhf(v.z + b[2]);
  float h3 = tanhf(v.w + b[3]);
  float2 o;
  o.x = h0 * W[0] + h1 * W[2] + h2 * W[4] + h3 * W[6];
  o.y = h0 * W[1] + h1 * W[3] + h2 * W[5] + h3 * W[7];
  reinterpret_cast<float2*>(out)[i] = o;
}

// emb = tanh(agg + b3) [N,2]; result = emb @ Wc (2x8) + bc
// d_out layout: [0, N*8) = result, [N*8, N*8 + N*2) = emb
__global__ void k_final(const float* __restrict__ agg, const float* __restrict__ b3,
                        const float* __restrict__ Wc, const float* __restrict__ bc,
                        float* __restrict__ out, int n) {
  int i = blockIdx.x * blockDim.x + threadIdx.x;
  if (i >= n) return;
  float2 v = reinterpret_cast<const float2*>(agg)[i];
  float e0 = tanhf(v.x + b3[0]);
  float e1 = tanhf(v.y + b3[1]);
  reinterpret_cast<float2*>(out + (size_t)n * 8)[i] = make_float2(e0, e1);
  float4 r0, r1;
  r0.x = e0 * Wc[0] + e1 * Wc[8]  + bc[0];
  r0.y = e0 * Wc[1] + e1 * Wc[9]  + bc[1];
  r0.z = e0 * Wc[2] + e1 * Wc[10] + bc[2];
  r0.w = e0 * Wc[3] + e1 * Wc[11] + bc[3];
  r1.x = e0 * Wc[4] + e1 * Wc[12] + bc[4];
  r1.y = e0 * Wc[5] + e1 * Wc[13] + bc[5];
  r1.z = e0 * Wc[6] + e1 * Wc[14] + bc[6];
  r1.w = e0 * Wc[7] + e1 * Wc[15] + bc[7];
  reinterpret_cast<float4*>(out)[(size_t)i * 2 + 0] = r0;
  reinterpret_cast<float4*>(out)[(size_t)i * 2 + 1] = r1;
}

// -------------------- host-side orchestration --------------------
extern "C" void kernel_launch(void* const* d_in, const int* in_sizes, int n_in,
                              void* d_out, int out_size, void* d_ws, size_t ws_size,
                              hipStream_t stream) {
  const float* x  = (const float*)d_in[0];
  const int*   ei = (const int*)d_in[1];
  const float* W1 = (const float*)d_in[2];
  const float* b1 = (const float*)d_in[3];
  const float* W2 = (const float*)d_in[4];
  const float* b2 = (const float*)d_in[5];
  const float* W3 = (const float*)d_in[6];
  const float* b3 = (const float*)d_in[7];
  const float* Wc = (const float*)d_in[8];
  const float* bc = (const float*)d_in[9];

  const int N = in_sizes[0] / 128;  // 500000
  const int E = in_sizes[1] / 2;    // 16000000
  const int* src = ei;
  const int* dst = ei + E;

  // workspace: dis[N] | buf0[N*4] | buf1[N*4]   (all 16B-aligned: N*4 % 16 == 0)
  float* dis  = (float*)d_ws;
  float* buf0 = dis + N;
  float* buf1 = buf0 + (size_t)N * 4;
  float* out  = (float*)d_out;

  const int T = 256;
  const int gN = cdiv(N, T);
  const int gE = cdiv(E, T);

  // 1) symmetric normalization: dis = rsqrt(1 + in-degree)
  k_deg_init<<<gN, T, 0, stream>>>(dis, N);
  k_deg_accum<<<gE, T, 0, stream>>>(dst, dis, E);
  k_deg_to_dis<<<gN, T, 0, stream>>>(dis, N);

  // 2) layer 1 dense transform via WMMA: buf0 = x @ W1
  const int wavesPerBlock = 4;  // 128 threads
  const int nWaves = cdiv(N, 16);
  k_gemm1_wmma<<<cdiv(nWaves, wavesPerBlock), wavesPerBlock * 32, 0, stream>>>(x, W1, buf0, N);

  // 3) layer 1 aggregation + fused tanh/W2: buf1 = agg(buf0); buf0 = tanh(buf1+b1)@W2
  k_self4<<<gN, T, 0, stream>>>(dis, buf0, buf1, N);
  k_edge4<<<gE, T, 0, stream>>>(src, dst, dis, buf0, buf1, E);
  k_act_gemm4x4<<<gN, T, 0, stream>>>(buf1, b1, W2, buf0, N);

  // 4) layer 2 aggregation + fused tanh/W3: buf0 -> [N,2]
  k_self4<<<gN, T, 0, stream>>>(dis, buf0, buf1, N);
  k_edge4<<<gE, T, 0, stream>>>(src, dst, dis, buf0, buf1, E);
  k_act_gemm4x2<<<gN, T, 0, stream>>>(buf1, b2, W3, buf0, N);

  // 5) layer 3 aggregation + fused tanh / classifier
  k_self2<<<gN, T, 0, stream>>>(dis, buf0, buf1, N);
  k_edge2<<<gE, T, 0, stream>>>(src, dst, dis, buf0, buf1, E);
  k_final<<<gN, T, 0, stream>>>(buf1, b3, Wc, bc, out, N);
}